// RoIRingPool_36069135351862
// MI455X (gfx1250) — compile-verified
//
#include <hip/hip_runtime.h>

// RoI Ring Pool for MI455X (gfx1250, wave32).
// features: [1,256,50,50] f32   rois: [2048,4] f32 (x1,y1,x2,y2)
// out:      [2048,256,7,7] f32
//
// One workgroup (256 threads = 8 waves) per RoI. Geometry is block-uniform
// (runs on the SALU float pipe, zero divergence). lane = channel; features
// (2.56 MB) stay L2-resident. The [256,49] tile is built in LDS and written
// back coalesced via GLOBAL_STORE_ASYNC_FROM_LDS_B128 (ASYNCcnt), drained
// with s_wait_asynccnt. Cell loops kept ROLLED to stay I$-resident.

#define C_CH   256
#define H_F    50
#define W_F    50
#define PH     7
#define PW     7
#define NCELL  (PH * PW)           // 49
#define TILE_F (C_CH * NCELL)      // 12544 floats = 50176 bytes
#define NCHUNK (TILE_F / 4)        // 3136 x 16B chunks
#define SPSCALE 0.0625f
#define SC_OUT  1.8f
#define NEGF   (-3.0e38f)

#if defined(__has_builtin)
#if __has_builtin(__builtin_amdgcn_global_store_async_from_lds_b128)
#define HAVE_ASYNC_STORE 1
#endif
#endif

// v4i32 pointers in explicit address spaces, matching the builtin signature:
//   void __builtin_amdgcn_global_store_async_from_lds_b128(
//       v4i32 as(1)* gaddr, v4i32 as(3)* ldsaddr, imm int offset, imm int cpol)
typedef int v4i __attribute__((ext_vector_type(4)));
typedef __attribute__((address_space(1))) v4i as1_v4i;
typedef __attribute__((address_space(3))) v4i as3_v4i;

__device__ __forceinline__ int clampi(int v, int lo, int hi) {
    return v < lo ? lo : (v > hi ? hi : v);
}

__global__ __launch_bounds__(256) void roi_ring_pool_kernel(
    const float* __restrict__ feat,   // [256,50,50]
    const float* __restrict__ rois,   // [N,4]
    float* __restrict__ out) {        // [N,256,7,7]
    __shared__ __align__(16) float tile[TILE_F];

    const int n = blockIdx.x;
    const int c = threadIdx.x;        // lane/thread = channel

    // ---- block-uniform RoI geometry (matches jnp: round = rne, floor/ceil) ----
    const float x1 = rois[n * 4 + 0];
    const float y1 = rois[n * 4 + 1];
    const float x2 = rois[n * 4 + 2];
    const float y2 = rois[n * 4 + 3];
    const float cx = (x1 + x2) * 0.5f, cy = (y1 + y2) * 0.5f;
    const float whf = (x2 - x1) * 0.5f, hhf = (y2 - y1) * 0.5f;

    // outer box (scaled 1.8x about center), feature coords, round-half-even
    const int rsw = (int)rintf((cx - whf * SC_OUT) * SPSCALE);
    const int rsh = (int)rintf((cy - hhf * SC_OUT) * SPSCALE);
    const int rew = (int)rintf((cx + whf * SC_OUT) * SPSCALE);
    const int reh = (int)rintf((cy + hhf * SC_OUT) * SPSCALE);
    // inner box (scale_inner = 1.0 -> original roi)
    const int isw = (int)rintf(x1 * SPSCALE);
    const int ish = (int)rintf(y1 * SPSCALE);
    const int iew = (int)rintf(x2 * SPSCALE);
    const int ieh = (int)rintf(y2 * SPSCALE);

    int rw = rew - rsw + 1; if (rw < 1) rw = 1;
    int rh = reh - rsh + 1; if (rh < 1) rh = 1;
    const float bw = (float)rw / (float)PW;
    const float bh = (float)rh / (float)PH;

    const float* featC = feat + (size_t)c * (H_F * W_F);
    // warm L2/L0 for this channel's region (global_prefetch_b8)
    __builtin_prefetch(featC + clampi(rsh, 0, H_F - 1) * W_F, 0, 1);

    // ---- ring max-pool: bin minus strict interior of inner box ----
    // Keep the 7x7 cell loops ROLLED: inner (h,w) loops are dynamic anyway,
    // and a rolled body keeps the whole kernel instruction-cache resident.
    #pragma clang loop unroll(disable)
    for (int ph = 0; ph < PH; ++ph) {
        const int hs = clampi((int)floorf((float)ph * bh) + rsh, 0, H_F);
        const int he = clampi((int)ceilf((float)(ph + 1) * bh) + rsh, 0, H_F);
        #pragma clang loop unroll(disable)
        for (int pw = 0; pw < PW; ++pw) {
            const int ws = clampi((int)floorf((float)pw * bw) + rsw, 0, W_F);
            const int we = clampi((int)ceilf((float)(pw + 1) * bw) + rsw, 0, W_F);
            float m = NEGF;
            bool found = false;   // block-uniform -> scalar flag
            for (int h = hs; h < he; ++h) {
                const bool inh = (h > ish) && (h < ieh);
                const float* row = featC + h * W_F;
                for (int w = ws; w < we; ++w) {
                    const bool inw = (w > isw) && (w < iew);
                    if (!(inh && inw)) {
                        m = fmaxf(m, row[w]);
                        found = true;
                    }
                }
            }
            tile[c * NCELL + ph * PW + pw] = found ? m : 0.0f;
        }
    }

    __syncthreads();

    // ---- coalesced tile writeback: 3136 x B128 chunks (50176 B) ----
    float* gbase = out + (size_t)n * TILE_F;   // 16B-aligned (50176 % 16 == 0)
#if defined(HAVE_ASYNC_STORE)
    #pragma clang loop unroll(disable)
    for (int idx = c; idx < NCHUNK; idx += 256) {
        __builtin_amdgcn_global_store_async_from_lds_b128(
            (as1_v4i*)(gbase + idx * 4), (as3_v4i*)(tile + idx * 4),
            /*offset=*/0, /*cpol=*/0);
    }
#if __has_builtin(__builtin_amdgcn_s_wait_asynccnt)
    __builtin_amdgcn_s_wait_asynccnt(0);
#else
    asm volatile("s_wait_asynccnt 0" ::: "memory");
#endif
#else
    #pragma clang loop unroll(disable)
    for (int idx = c; idx < NCHUNK; idx += 256) {
        float4 v = *(const float4*)(tile + idx * 4);
        *(float4*)(gbase + idx * 4) = v;
    }
#endif
}

extern "C" void kernel_launch(void* const* d_in, const int* in_sizes, int n_in,
                              void* d_out, int out_size, void* d_ws, size_t ws_size,
                              hipStream_t stream) {
    const float* feat = (const float*)d_in[0];   // [1,256,50,50]
    const float* rois = (const float*)d_in[1];   // [N,4]
    float* out = (float*)d_out;                  // [N,256,7,7]
    const int N = in_sizes[1] / 4;               // 2048
    roi_ring_pool_kernel<<<dim3(N), dim3(256), 0, stream>>>(feat, rois, out);
}